// CompressImageGPU_41377714930159
// MI455X (gfx1250) — compile-verified
//
#include <hip/hip_runtime.h>
#include <stdint.h>

// CompressImageGPU for MI455X (gfx1250):
//   y = clip( A_h * X * A_w^T + blocknoise , -1, 1 )
// A = (bilinear up x2) o (antialiased bilinear down x0.5): banded 512x512.
// Per-16x16-tile f32 WMMA GEMMs (V_WMMA_F32_16X16X4_F32); interior band
// weights hoisted to one per-wave register set; async global->LDS staging
// (global_load_async_to_lds_b128/b32, ASYNCcnt).

typedef __attribute__((ext_vector_type(2))) float v2f;
typedef __attribute__((ext_vector_type(8))) float v8f;

#define HH 512
#define WW 512

// Antialiased downsample row k of D (512->256), jax truncated+renormalized edges.
__device__ __forceinline__ float Dw(int k, int j) {
  if (k == 0) {
    if (j == 0 || j == 1) return 3.0f / 7.0f;
    if (j == 2) return 1.0f / 7.0f;
    return 0.0f;
  }
  if (k == 255) {
    if (j == 510 || j == 511) return 3.0f / 7.0f;
    if (j == 509) return 1.0f / 7.0f;
    return 0.0f;
  }
  int o = j - (2 * k - 1);  // interior: {1,3,3,1}/8
  if (o == 0 || o == 3) return 0.125f;
  if (o == 1 || o == 2) return 0.375f;
  return 0.0f;
}

// Composed operator A[i][j] = sum_k U[i][k] * D[k][j]
__device__ __forceinline__ float opA(int i, int j) {
  if (i == 0)   return Dw(0, j);
  if (i == 511) return Dw(255, j);
  int k = i >> 1;
  if (i & 1) return 0.75f * Dw(k, j) + 0.25f * Dw(k + 1, j);
  return 0.25f * Dw(k - 1, j) + 0.75f * Dw(k, j);
}

__global__ __launch_bounds__(256) void CompressImageGPU_kernel(
    const float* __restrict__ x, const float* __restrict__ bn,
    float* __restrict__ out) {
  // 72 rows x 76 cols halo region (base gr0-3 / gc0-4); stride 76 -> column
  // reads conflict-free (12*m mod 64 distinct for m<16; lane-halves disjoint).
  __shared__ __align__(16) float sx[72][76];
  // Per-wave stage-1 spill; stride 18 keeps the two lane-halves on disjoint banks.
  __shared__ float sS[8][32][18];

  const int tid = threadIdx.x;
  const int lane = tid & 31;  // wave32
  const int wv = tid >> 5;    // 8 waves
  const int wg = blockIdx.x;
  const int gc0 = (wg & 7) * 64;
  const int gr0 = ((wg >> 3) & 7) * 64;
  const int bc = wg >> 6;  // b*3 + c
  const size_t xbase = (size_t)bc * (HH * WW);

  // ---- Stage 0: async global->LDS staging of the halo region ---------------
  // Clamped/zero halo values are only ever hit by zero weights (edge rows of A
  // use renormalized in-range taps), so this is exact.
  const bool colEdge = (gc0 == 0) || (gc0 == WW - 64);
  if (!colEdge) {
    // Vector path: 72 rows x 19 float4; base col gc0-4 is 16B-aligned; only a
    // row clamp is needed, which preserves 16B contiguity.
    for (int idx = tid; idx < 72 * 19; idx += 256) {
      int r = idx / 19;
      int q = idx - r * 19;
      int gr = gr0 - 3 + r; gr = gr < 0 ? 0 : (gr > HH - 1 ? HH - 1 : gr);
      const float* gp = x + xbase + (size_t)gr * WW + (gc0 - 4) + 4 * q;
      unsigned ldsa = (unsigned)(uintptr_t)&sx[r][4 * q];
      asm volatile("global_load_async_to_lds_b128 %0, %1, off"
                   :: "v"(ldsa), "v"(gp) : "memory");
    }
  } else {
    // Edge-column workgroups: per-element clamp, b32 async copies.
    for (int idx = tid; idx < 72 * 76; idx += 256) {
      int r = idx / 76;
      int c = idx - r * 76;
      int gr = gr0 - 3 + r; gr = gr < 0 ? 0 : (gr > HH - 1 ? HH - 1 : gr);
      int gc = gc0 - 4 + c; gc = gc < 0 ? 0 : (gc > WW - 1 ? WW - 1 : gc);
      const float* gp = x + xbase + (size_t)gr * WW + gc;
      unsigned ldsa = (unsigned)(uintptr_t)&sx[r][c];
      asm volatile("global_load_async_to_lds_b32 %0, %1, off"
                   :: "v"(ldsa), "v"(gp) : "memory");
    }
  }
  asm volatile("s_wait_asynccnt 0x0" ::: "memory");
  __syncthreads();

  // V_WMMA_F32_16X16X4_F32 lane decomposition:
  //   A 16x4 : lanes 0-15 -> M=lane, K={0,1}; lanes 16-31 -> K={2,3}
  //   B 4x16 : lanes hold N; VGPR{0,1} x lane-half hold K={0,1}/{2,3}
  //   C/D    : VGPR j -> row j (lanes 0-15) / row j+8 (lanes 16-31)
  const int n = lane & 15;
  const int half = lane >> 4;
  const int kh = half * 2;

  // Interior band weights: identical per-lane values serve stage-1 B-operand
  // (Bw[k][n] = A[c0+n][c0-3+k]) and stage-2 A-operand (Ah[m][k] =
  // A[r0+m][r0-3+k]) for every non-edge tile. Computed once per wave.
  v2f WI[6];
#pragma unroll
  for (int q = 0; q < 6; ++q) {
    WI[q].x = opA(16 + n, 13 + 4 * q + kh);
    WI[q].y = opA(16 + n, 14 + 4 * q + kh);
  }

  for (int t = wv; t < 16; t += 8) {  // 2 tiles of 16x16 per wave, uniform
    const int r0 = gr0 + (t >> 2) * 16;
    const int c0 = gc0 + (t & 3) * 16;
    int rstart = r0 - 3; rstart = rstart < 0 ? 0 : (rstart > HH - 24 ? HH - 24 : rstart);
    int kx = c0 - 3;     kx = kx < 0 ? 0 : (kx > WW - 24 ? WW - 24 : kx);
    const int rl0 = rstart - (gr0 - 3);
    const int kxl = kx - (gc0 - 4);

    v2f WB[6], WA[6];
#pragma unroll
    for (int q = 0; q < 6; ++q) { WB[q] = WI[q]; WA[q] = WI[q]; }
    if (c0 == 0 || c0 == WW - 16) {  // wave-uniform branch (EXEC untouched)
#pragma unroll
      for (int q = 0; q < 6; ++q) {
        WB[q].x = opA(c0 + n, kx + 4 * q + kh);
        WB[q].y = opA(c0 + n, kx + 4 * q + kh + 1);
      }
    }
    if (r0 == 0 || r0 == HH - 16) {
#pragma unroll
      for (int q = 0; q < 6; ++q) {
        WA[q].x = opA(r0 + n, rstart + 4 * q + kh);
        WA[q].y = opA(r0 + n, rstart + 4 * q + kh + 1);
      }
    }

    // ---- Stage 1: S(32x16) = Xwin(32x24) * Bw(24x16) -----------------------
    v8f s0 = {0.f, 0.f, 0.f, 0.f, 0.f, 0.f, 0.f, 0.f};
    v8f s1 = {0.f, 0.f, 0.f, 0.f, 0.f, 0.f, 0.f, 0.f};
#pragma unroll
    for (int q = 0; q < 6; ++q) {
      const int kq = 4 * q + kh;
      const int ra = rl0 + n;                         // rows rstart..rstart+15
      int rb = rl0 + 16 + n; rb = rb > 71 ? 71 : rb;  // rows 24..31 junk, never read
      v2f a0; a0.x = sx[ra][kxl + kq]; a0.y = sx[ra][kxl + kq + 1];
      v2f a1; a1.x = sx[rb][kxl + kq]; a1.y = sx[rb][kxl + kq + 1];
      s0 = __builtin_amdgcn_wmma_f32_16x16x4_f32(false, a0, false, WB[q],
                                                 (short)0, s0, false, false);
      s1 = __builtin_amdgcn_wmma_f32_16x16x4_f32(false, a1, false, WB[q],
                                                 (short)0, s1, false, false);
    }
    // Spill D-layout -> row-major S in wave-private LDS (same-wave DS ordering)
#pragma unroll
    for (int j = 0; j < 8; ++j) {
      sS[wv][j + 8 * half][n] = s0[j];
      sS[wv][16 + j + 8 * half][n] = s1[j];
    }

    // ---- Stage 2: Y(16x16) = Ah(16x24) * S(24x16) --------------------------
    v8f y = {0.f, 0.f, 0.f, 0.f, 0.f, 0.f, 0.f, 0.f};
#pragma unroll
    for (int q = 0; q < 6; ++q) {
      const int kq = 4 * q + kh;
      v2f bs;
      bs.x = sS[wv][kq][n];
      bs.y = sS[wv][kq + 1][n];
      y = __builtin_amdgcn_wmma_f32_16x16x4_f32(false, WA[q], false, bs,
                                                (short)0, y, false, false);
    }

    // ---- Fused noise + clamp + store ---------------------------------------
    const int gcol = c0 + n;
    const int rblk = (r0 + 8 * half) >> 3;  // constant over j (r0 % 16 == 0)
    const float nz =
        (bn[((size_t)(bc / 3) * 64 + rblk) * 64 + (gcol >> 3)] - 0.5f) * 0.02f;
#pragma unroll
    for (int j = 0; j < 8; ++j) {
      const int grow = r0 + 8 * half + j;
      float v = fminf(1.0f, fmaxf(-1.0f, y[j] + nz));
      out[xbase + (size_t)grow * WW + gcol] = v;
    }
  }
}

extern "C" void kernel_launch(void* const* d_in, const int* in_sizes, int n_in,
                              void* d_out, int out_size, void* d_ws,
                              size_t ws_size, hipStream_t stream) {
  (void)in_sizes; (void)n_in; (void)out_size; (void)d_ws; (void)ws_size;
  const float* x = (const float*)d_in[0];   // [32,3,512,512] f32
  const float* bn = (const float*)d_in[1];  // [32,64,64] f32
  float* out = (float*)d_out;               // [32,3,512,512] f32
  const int nblocks = 32 * 3 * (HH / 64) * (WW / 64);  // 6144
  CompressImageGPU_kernel<<<dim3(nblocks), dim3(256), 0, stream>>>(x, bn, out);
}